// SemLoss_45990509805976
// MI455X (gfx1250) — compile-verified
//
#include <hip/hip_runtime.h>
#include <hip/hip_bf16.h>
#include <stdint.h>

// ---------------- configuration ----------------
#define CLS    20
#define BINS   4096
#define NBLK1  1024
#define CHUNK  16384
#define ESCALE 131072.0f   // 2^17 fixed-point scale for per-bin error sums

typedef __attribute__((ext_vector_type(2))) float v2f;
typedef __attribute__((ext_vector_type(8))) float v8f;

// workspace layout (bytes)
#define OFF_CNT   0u          // uint  [20*4096]
#define OFF_FG    327680u     // uint  [20*4096]
#define OFF_ES    655360u     // u64   [20*4096]
#define OFF_CEPS  1310720u    // float [NBLK1]
#define OFF_CEPC  1314816u    // uint  [NBLK1]
#define OFF_CLOSS 1318912u    // double[20]
#define OFF_PRES  1319072u    // uint  [20]
#define OFF_META  1376256u    // float [P]   (log-partition per point)
#define ZERO_BYTES 1310720u   // only the atomic histogram region needs zeroing

#if defined(__gfx1250__) && __has_builtin(__builtin_amdgcn_wmma_f32_16x16x4_f32)
#define USE_WMMA44 1
#endif

// Swap the two 16-lane halves of a wave32 (lane L <-> lane L^16).
// v_permlanex16_b32 with identity selects = pure VALU, no LDS traffic,
// no s_wait_dscnt stall (unlike __shfl_xor -> ds_bpermute_b32).
__device__ __forceinline__ float xor16f(float v) {
#if defined(__gfx1250__) && __has_builtin(__builtin_amdgcn_permlanex16)
    int i = __float_as_int(v);
    int r = __builtin_amdgcn_permlanex16(i, i, 0x76543210, (int)0xfedcba98u, false, false);
    return __int_as_float(r);
#else
    return __shfl_xor(v, 16, 32);
#endif
}

// ---------------------------------------------------------------------------
// Pass 1: fused log-softmax. Each pair of lanes (m, m+16) owns point m of a
// 16-point tile; lane half `hi` holds classes 4j+2*hi, 4j+2*hi+1. The 20-class
// sum of exp is computed with 5 chained v_wmma_f32_16x16x4_f32 (B = ones,
// which makes the row-sum independent of the A/B K-lane mapping).
// Emits: meta[p] = logZ, per-block CE partial sums (deterministic).
// ---------------------------------------------------------------------------
__global__ __launch_bounds__(256) void k_softmax_ce(
    const float* __restrict__ xin, const int* __restrict__ tgt,
    float* __restrict__ meta, float* __restrict__ cePS,
    unsigned int* __restrict__ cePC, int P)
{
    const int lane = threadIdx.x & 31;
    const int wave = threadIdx.x >> 5;
    const int hi   = lane >> 4;        // which half of the pair
    const int m    = lane & 15;        // point-in-tile == matrix row M
    const int tiles = P >> 4;
    const int gw = blockIdx.x * 8 + wave;
    const int nw = gridDim.x * 8;

    float ceS = 0.f; unsigned int ceC = 0;
    const v2f bones = {1.0f, 1.0f};
    const int cbase = 2 * hi;          // first class of this lane's pair slots

    for (int t = gw; t < tiles; t += nw) {
        const int p = t * 16 + m;
        const float2* row = (const float2*)(xin + (size_t)p * CLS);
        float2 x0 = row[0 + hi], x1 = row[2 + hi], x2 = row[4 + hi],
               x3 = row[6 + hi], x4 = row[8 + hi];

        // row max across both lanes of the pair (wave32 half-swap)
        float mx = fmaxf(fmaxf(fmaxf(x0.x, x0.y), fmaxf(x1.x, x1.y)),
                   fmaxf(fmaxf(x2.x, x2.y),
                   fmaxf(fmaxf(x3.x, x3.y), fmaxf(x4.x, x4.y))));
        mx = fmaxf(mx, xor16f(mx));

        v2f e0 = {__expf(x0.x - mx), __expf(x0.y - mx)};
        v2f e1 = {__expf(x1.x - mx), __expf(x1.y - mx)};
        v2f e2 = {__expf(x2.x - mx), __expf(x2.y - mx)};
        v2f e3 = {__expf(x3.x - mx), __expf(x3.y - mx)};
        v2f e4 = {__expf(x4.x - mx), __expf(x4.y - mx)};

        float S;
#ifdef USE_WMMA44
        v8f acc = {0.f,0.f,0.f,0.f,0.f,0.f,0.f,0.f};
        acc = __builtin_amdgcn_wmma_f32_16x16x4_f32(false, e0, false, bones, (short)0, acc, false, false);
        acc = __builtin_amdgcn_wmma_f32_16x16x4_f32(false, e1, false, bones, (short)0, acc, false, false);
        acc = __builtin_amdgcn_wmma_f32_16x16x4_f32(false, e2, false, bones, (short)0, acc, false, false);
        acc = __builtin_amdgcn_wmma_f32_16x16x4_f32(false, e3, false, bones, (short)0, acc, false, false);
        acc = __builtin_amdgcn_wmma_f32_16x16x4_f32(false, e4, false, bones, (short)0, acc, false, false);
        // D layout: vgpr r = row M=r (lanes 0-15) / M=r+8 (lanes 16-31); B=ones
        // makes every column identical. 8->1 mux on low lane bits, then pair swap.
        {
            float t0 = (lane & 1) ? acc[1] : acc[0];
            float t1 = (lane & 1) ? acc[3] : acc[2];
            float t2 = (lane & 1) ? acc[5] : acc[4];
            float t3 = (lane & 1) ? acc[7] : acc[6];
            float u0 = (lane & 2) ? t1 : t0;
            float u1 = (lane & 2) ? t3 : t2;
            float sOwn = (lane & 4) ? u1 : u0;
            float sPart = xor16f(sOwn);
            bool ownOk = (((lane >> 3) & 1) == ((lane >> 4) & 1));
            S = ownOk ? sOwn : sPart;
        }
#else
        float sp = e0.x+e0.y+e1.x+e1.y+e2.x+e2.y+e3.x+e3.y+e4.x+e4.y;
        S = sp + xor16f(sp);
#endif
        float logZ = mx + __logf(S);

        int tg = tgt[p];
        // x_t gather: flat predicated selects (no branches / EXEC games).
        // This lane holds classes 4j + cbase + {0,1}, j = 0..4.
        float cand = -3.4e38f;
        cand = (tg == ( 0 + cbase    )) ? x0.x : cand;
        cand = (tg == ( 0 + cbase + 1)) ? x0.y : cand;
        cand = (tg == ( 4 + cbase    )) ? x1.x : cand;
        cand = (tg == ( 4 + cbase + 1)) ? x1.y : cand;
        cand = (tg == ( 8 + cbase    )) ? x2.x : cand;
        cand = (tg == ( 8 + cbase + 1)) ? x2.y : cand;
        cand = (tg == (12 + cbase    )) ? x3.x : cand;
        cand = (tg == (12 + cbase + 1)) ? x3.y : cand;
        cand = (tg == (16 + cbase    )) ? x4.x : cand;
        cand = (tg == (16 + cbase + 1)) ? x4.y : cand;
        float xt = fmaxf(cand, xor16f(cand));

        if (hi == 0) {
            meta[p] = logZ;
            if (tg != 0) { ceS += (logZ - xt); ceC += 1u; }
        }
    }

    // scalar tail (P not multiple of 16) — inactive for P = 2,000,000
    int rem = P & 15;
    int gtid = blockIdx.x * 256 + threadIdx.x;
    if (gtid < rem) {
        int p = (P & ~15) + gtid;
        float mx = -3.4e38f;
        for (int c = 0; c < CLS; ++c) mx = fmaxf(mx, xin[(size_t)p * CLS + c]);
        float s = 0.f;
        for (int c = 0; c < CLS; ++c) s += __expf(xin[(size_t)p * CLS + c] - mx);
        float logZ = mx + __logf(s);
        meta[p] = logZ;
        int tg = tgt[p];
        if (tg != 0) { ceS += (logZ - xin[(size_t)p * CLS + tg]); ceC += 1u; }
    }

    __shared__ float sS[256];
    __shared__ unsigned int sC[256];
    sS[threadIdx.x] = ceS; sC[threadIdx.x] = ceC; __syncthreads();
    for (int o = 128; o > 0; o >>= 1) {
        if (threadIdx.x < o) { sS[threadIdx.x] += sS[threadIdx.x + o]; sC[threadIdx.x] += sC[threadIdx.x + o]; }
        __syncthreads();
    }
    if (threadIdx.x == 0) { cePS[blockIdx.x] = sS[0]; cePC[blockIdx.x] = sC[0]; }
}

// ---------------------------------------------------------------------------
// Pass 2: per-(class, point-chunk) error histogram in LDS, merged to global
// with integer atomics (deterministic). Input re-read is L2-resident (160MB
// fits in the 192MB L2). e = |fg - softmax_c| in [0,1], uniform 4096 bins.
// ---------------------------------------------------------------------------
__global__ __launch_bounds__(256) void k_hist(
    const float* __restrict__ xin, const int* __restrict__ tgt,
    const float* __restrict__ meta,
    unsigned int* __restrict__ gCnt, unsigned int* __restrict__ gFg,
    unsigned long long* __restrict__ gEs, int P)
{
    __shared__ unsigned int hC[BINS], hF[BINS], hE[BINS];
    for (int i = threadIdx.x; i < BINS; i += 256) { hC[i] = 0u; hF[i] = 0u; hE[i] = 0u; }
    __syncthreads();

    const int c  = blockIdx.y;
    const int p0 = blockIdx.x * CHUNK;
    for (int i = threadIdx.x; i < CHUNK; i += 256) {
        int p = p0 + i;
        if (p >= P) break;
        __builtin_prefetch(&xin[(size_t)(p + 1024) * CLS + c], 0, 1);  // global_prefetch_b8
        float x  = xin[(size_t)p * CLS + c];
        float lz = meta[p];
        int   tg = tgt[p];
        int   fg = (tg == c);
        float pc = __expf(x - lz);
        float e  = fg ? (1.f - pc) : pc;
        e = fminf(fmaxf(e, 0.f), 1.f);
        int b = (int)(e * (float)BINS); if (b > BINS - 1) b = BINS - 1;
        atomicAdd(&hC[b], 1u);
        if (fg) atomicAdd(&hF[b], 1u);
        atomicAdd(&hE[b], (unsigned int)(e * ESCALE + 0.5f));   // <= 16384*2^17 < 2^32
    }
    __syncthreads();

    unsigned int* gc = gCnt + c * BINS;
    unsigned int* gf = gFg  + c * BINS;
    unsigned long long* ge = gEs + c * BINS;
    for (int i = threadIdx.x; i < BINS; i += 256) {
        unsigned int cc = hC[i];
        if (cc) {
            atomicAdd(&gc[i], cc);
            unsigned int ff = hF[i]; if (ff) atomicAdd(&gf[i], ff);
            unsigned int ee = hE[i]; if (ee) atomicAdd(&ge[i], (unsigned long long)ee);
        }
    }
}

// ---------------------------------------------------------------------------
// Pass 3: per class, descending-e prefix scan over bins -> Lovász loss.
// (cnt,fg) packed in one u64 and scanned together; jaccard in fp64.
// ---------------------------------------------------------------------------
__global__ __launch_bounds__(256) void k_lovasz(
    const unsigned int* __restrict__ gCnt, const unsigned int* __restrict__ gFg,
    const unsigned long long* __restrict__ gEs,
    double* __restrict__ closs, unsigned int* __restrict__ pres)
{
    const int c = blockIdx.x;
    const int t = threadIdx.x;
    const unsigned int* gc = gCnt + c * BINS;
    const unsigned int* gf = gFg  + c * BINS;
    const unsigned long long* ge = gEs + c * BINS;

    unsigned int lc[16], lf[16];
    unsigned long long tot = 0ull;
#pragma unroll
    for (int q = 0; q < 16; ++q) {
        int b = BINS - 1 - (t * 16 + q);     // descending e order
        lc[q] = gc[b]; lf[q] = gf[b];
        tot += ((unsigned long long)lc[q] << 32) | (unsigned long long)lf[q];
    }

    __shared__ unsigned long long ss[256];
    ss[t] = tot; __syncthreads();
    for (int o = 1; o < 256; o <<= 1) {
        unsigned long long v = (t >= o) ? ss[t - o] : 0ull;
        __syncthreads();
        ss[t] += v;
        __syncthreads();
    }
    unsigned long long excl  = ss[t] - tot;
    unsigned long long total = ss[255];
    unsigned int G = (unsigned int)(total & 0xffffffffull);

    double acc = 0.0;
    if (G > 0u) {
        long long n = (long long)(excl >> 32);
        long long k = (long long)(excl & 0xffffffffull);
#pragma unroll
        for (int q = 0; q < 16; ++q) {
            unsigned int cc = lc[q];
            if (cc) {
                int b = BINS - 1 - (t * 16 + q);
                double em = (double)ge[b] * (1.0 / 131072.0) / (double)cc;
                double js = 1.0 - (double)((long long)G - k) / (double)((long long)G + n - k);
                n += (long long)cc; k += (long long)lf[q];
                double je = 1.0 - (double)((long long)G - k) / (double)((long long)G + n - k);
                acc += em * (je - js);
            }
        }
    }

    __shared__ double sr[256];
    sr[t] = acc; __syncthreads();
    for (int o = 128; o > 0; o >>= 1) {
        if (t < o) sr[t] += sr[t + o];
        __syncthreads();
    }
    if (t == 0) { closs[c] = (G > 0u) ? sr[0] : 0.0; pres[c] = (G > 0u) ? 1u : 0u; }
}

// ---------------------------------------------------------------------------
// Pass 4: final combine (fixed-order, deterministic).
// ---------------------------------------------------------------------------
__global__ __launch_bounds__(128) void k_final(
    const float* __restrict__ cePS, const unsigned int* __restrict__ cePC,
    const double* __restrict__ closs, const unsigned int* __restrict__ pres,
    float* __restrict__ outv, int nblk)
{
    int t = threadIdx.x;
    double s = 0.0; unsigned long long cn = 0ull;
    for (int i = t; i < nblk; i += 128) { s += (double)cePS[i]; cn += cePC[i]; }
    double lv = 0.0; unsigned int np = 0u;
    if (t < CLS) { lv = closs[t]; np = pres[t]; }

    __shared__ double sd[128]; __shared__ unsigned long long sc[128];
    __shared__ double sl[128]; __shared__ unsigned int sp[128];
    sd[t] = s; sc[t] = cn; sl[t] = lv; sp[t] = np; __syncthreads();
    for (int o = 64; o > 0; o >>= 1) {
        if (t < o) { sd[t] += sd[t+o]; sc[t] += sc[t+o]; sl[t] += sl[t+o]; sp[t] += sp[t+o]; }
        __syncthreads();
    }
    if (t == 0) {
        double ce  = sd[0] / (double)(sc[0] ? sc[0] : 1ull);
        double lov = sl[0] / (double)(sp[0] ? sp[0] : 1u);
        outv[0] = (float)(2.0 * ce);
        outv[1] = (float)(6.0 * lov);
    }
}

// ---------------------------------------------------------------------------
extern "C" void kernel_launch(void* const* d_in, const int* in_sizes, int n_in,
                              void* d_out, int out_size, void* d_ws, size_t ws_size,
                              hipStream_t stream)
{
    const float* outputs = (const float*)d_in[0];
    const int*   targets = (const int*)d_in[1];
    const int P = in_sizes[1];

    char* ws = (char*)d_ws;
    unsigned int*       gCnt  = (unsigned int*)(ws + OFF_CNT);
    unsigned int*       gFg   = (unsigned int*)(ws + OFF_FG);
    unsigned long long* gEs   = (unsigned long long*)(ws + OFF_ES);
    float*              cePS  = (float*)(ws + OFF_CEPS);
    unsigned int*       cePC  = (unsigned int*)(ws + OFF_CEPC);
    double*             closs = (double*)(ws + OFF_CLOSS);
    unsigned int*       pres  = (unsigned int*)(ws + OFF_PRES);
    float*              meta  = (float*)(ws + OFF_META);

    hipMemsetAsync(d_ws, 0, ZERO_BYTES, stream);   // zero atomic accumulators each call

    k_softmax_ce<<<NBLK1, 256, 0, stream>>>(outputs, targets, meta, cePS, cePC, P);

    int chunks = (P + CHUNK - 1) / CHUNK;
    k_hist<<<dim3(chunks, CLS), 256, 0, stream>>>(outputs, targets, meta, gCnt, gFg, gEs, P);

    k_lovasz<<<CLS, 256, 0, stream>>>(gCnt, gFg, gEs, closs, pres);

    k_final<<<1, 128, 0, stream>>>(cePS, cePC, closs, pres, (float*)d_out, NBLK1);
}